// GCN_72215580115011
// MI455X (gfx1250) — compile-verified
//
#include <hip/hip_runtime.h>
#include <hip/hip_bf16.h>

typedef __attribute__((ext_vector_type(2))) float v2f;
typedef __attribute__((ext_vector_type(8))) float v8f;

#define FOUT 32

static __device__ __forceinline__ void atomic_add_f32(float* p, float v) {
    __hip_atomic_fetch_add(p, v, __ATOMIC_RELAXED, __HIP_MEMORY_SCOPE_AGENT);
}

// ---------------- zero fill ----------------
__global__ void zero_kernel(float* __restrict__ p, long n) {
    long i4 = ((long)blockIdx.x * blockDim.x + threadIdx.x) * 4;
    if (i4 + 3 < n) {
        *(float4*)(p + i4) = make_float4(0.f, 0.f, 0.f, 0.f);
    } else {
        for (long j = i4; j < n; ++j) p[j] = 0.f;
    }
}

// ---------------- degree accumulation ----------------
__global__ void degree_kernel(const int* __restrict__ src, const int* __restrict__ dst,
                              float* __restrict__ deg_out, float* __restrict__ deg_in, int E) {
    int e = blockIdx.x * blockDim.x + threadIdx.x;
    if (e >= E) return;
    atomic_add_f32(deg_out + src[e], 1.0f);
    atomic_add_f32(deg_in + dst[e], 1.0f);
}

// ---------------- deg -> rsqrt(clip(deg,1)) in place ----------------
__global__ void norm_kernel(float* __restrict__ d, int n) {
    int i = blockIdx.x * blockDim.x + threadIdx.x;
    if (i >= n) return;
    float v = d[i];
    d[i] = rsqrtf(v > 1.0f ? v : 1.0f);
}

// ---------------- WMMA GEMM: out[m, 0:32] = (A[m, 0:K] @ W[K, 32]) * rownorm[m]
// one wave (32 lanes) computes a 16x16 tile of the output via V_WMMA_F32_16X16X4_F32
__global__ void gemm_scale_kernel(const float* __restrict__ A, const float* __restrict__ W,
                                  int K, const float* __restrict__ rownorm,
                                  float* __restrict__ out, int N) {
    const int lane  = threadIdx.x;      // 0..31, one wave per block
    const int mtile = blockIdx.x;
    const int ntile = blockIdx.y;       // 0 or 1 (FOUT=32 -> two 16-col tiles)
    const int half  = lane >> 4;        // 0: lanes 0-15, 1: lanes 16-31
    const int l16   = lane & 15;
    const int kbase = half * 2;         // A/B K-pair selector per ISA layout

    int row = mtile * 16 + l16;
    if (row >= N) row = N - 1;          // clamp (grid is exact for N=100000)
    const int ncol = ntile * 16 + l16;

    const float* arow = A + (long)row * K;

    v8f c = {};
    for (int k = 0; k < K; k += 4) {
        // A 16x4 tile: lane holds A[m=l16][kbase], A[m=l16][kbase+1]
        v2f a = *(const v2f*)(arow + k + kbase);
        // B 4x16 tile (mirrored layout): lane holds W[k+kbase][ncol], W[k+kbase+1][ncol]
        v2f b;
        b.x = W[(k + kbase) * FOUT + ncol];
        b.y = W[(k + kbase + 1) * FOUT + ncol];
        c = __builtin_amdgcn_wmma_f32_16x16x4_f32(
                /*neg_a=*/false, a, /*neg_b=*/false, b,
                /*c_mod=*/(short)0, c, /*reuse_a=*/false, /*reuse_b=*/false);
    }

    // C/D layout: VGPR j holds row m = mtile*16 + j + 8*half, col = ncol
#pragma unroll
    for (int j = 0; j < 8; ++j) {
        int m = mtile * 16 + j + 8 * half;
        if (m < N) {
            out[(long)m * FOUT + ncol] = c[j] * rownorm[m];
        }
    }
}

// ---------------- edge scatter: AGG[dst] += H[src]  (8 threads/edge, float4 each) ----
__global__ void scatter_kernel(const float* __restrict__ H, const int* __restrict__ src,
                               const int* __restrict__ dst, float* __restrict__ AGG, int E) {
    long tid = (long)blockIdx.x * blockDim.x + threadIdx.x;
    int e = (int)(tid >> 3);
    int part = (int)(tid & 7);
    if (e >= E) return;
    int s = src[e];
    int d = dst[e];
    const float4 v = *(const float4*)(H + (long)s * FOUT + part * 4);
    float* o = AGG + (long)d * FOUT + part * 4;
    atomic_add_f32(o + 0, v.x);
    atomic_add_f32(o + 1, v.y);
    atomic_add_f32(o + 2, v.z);
    atomic_add_f32(o + 3, v.w);
}

// ---------------- epilogue: out = act(AGG * norm_in[row] + b[col]) -----------------
__global__ void epilogue_kernel(const float* __restrict__ AGG, const float* __restrict__ norm_in,
                                const float* __restrict__ b, float* __restrict__ out,
                                int N, int do_relu) {
    int tid = blockIdx.x * blockDim.x + threadIdx.x;   // N*32 <= 3.2M, fits int
    if (tid >= N * FOUT) return;
    int i = tid >> 5;
    int f = tid & 31;
    float v = AGG[tid] * norm_in[i] + b[f];
    if (do_relu) v = v > 0.f ? v : 0.f;
    out[tid] = v;
}

extern "C" void kernel_launch(void* const* d_in, const int* in_sizes, int n_in,
                              void* d_out, int out_size, void* d_ws, size_t ws_size,
                              hipStream_t stream) {
    const float* features = (const float*)d_in[0];   // [N,128]
    const float* W1       = (const float*)d_in[1];   // [128,32]
    const float* b1       = (const float*)d_in[2];   // [32]
    const float* W2       = (const float*)d_in[3];   // [32,32]
    const float* b2       = (const float*)d_in[4];   // [32]
    const int*   src      = (const int*)d_in[5];     // [E]
    const int*   dst      = (const int*)d_in[6];     // [E]

    const int FIN = 128;
    const int N = in_sizes[0] / FIN;                 // 100000
    const int E = in_sizes[5];                       // 1600000

    float* ws = (float*)d_ws;
    float* deg_out = ws;                             // N   (becomes norm_out in place)
    float* deg_in  = ws + N;                         // N   (becomes norm_in in place)
    float* H       = ws + 2 * (long)N;               // N*32 scaled projected features
    float* AGG     = H + (long)N * FOUT;             // N*32 aggregation buffer
    float* X1      = AGG + (long)N * FOUT;           // N*32 layer-1 activation

    const int BS = 256;
    const long degN = 2L * N;
    const long featN = (long)N * FOUT;

    // degrees + norms
    zero_kernel<<<(int)((degN / 4 + BS - 1) / BS), BS, 0, stream>>>(deg_out, degN);
    degree_kernel<<<(E + BS - 1) / BS, BS, 0, stream>>>(src, dst, deg_out, deg_in, E);
    norm_kernel<<<(int)((degN + BS - 1) / BS), BS, 0, stream>>>(deg_out, (int)degN);

    // ---- layer 1 ----
    dim3 ggrid((N + 15) / 16, 2);
    gemm_scale_kernel<<<ggrid, 32, 0, stream>>>(features, W1, FIN, deg_out, H, N);
    zero_kernel<<<(int)((featN / 4 + BS - 1) / BS), BS, 0, stream>>>(AGG, featN);
    scatter_kernel<<<(int)(((long)E * 8 + BS - 1) / BS), BS, 0, stream>>>(H, src, dst, AGG, E);
    epilogue_kernel<<<(int)((featN + BS - 1) / BS), BS, 0, stream>>>(AGG, deg_in, b1, X1, N, 1);

    // ---- layer 2 ----
    gemm_scale_kernel<<<ggrid, 32, 0, stream>>>(X1, W2, FOUT, deg_out, H, N);
    zero_kernel<<<(int)((featN / 4 + BS - 1) / BS), BS, 0, stream>>>(AGG, featN);
    scatter_kernel<<<(int)(((long)E * 8 + BS - 1) / BS), BS, 0, stream>>>(H, src, dst, AGG, E);
    epilogue_kernel<<<(int)((featN + BS - 1) / BS), BS, 0, stream>>>(AGG, deg_in, b2, (float*)d_out, N, 0);
}